// DeepFM_d_39427799777481
// MI455X (gfx1250) — compile-verified
//
#include <hip/hip_runtime.h>
#include <hip/hip_bf16.h>

// ---------------------------------------------------------------------------
// DeepFM forward, decomposed for MI455X (gfx1250, wave32, WMMA f32 16x16x4)
// ---------------------------------------------------------------------------
#define BZ 256
#define MM 100
#define NUF 16
#define NIF 32
#define EMB 128
#define DD 64
#define HH 128            // 2*DD
#define ROWS (BZ * MM)    // 25600
#define NF 17             // 1 + NUF fields for user / noise

typedef float v2f __attribute__((ext_vector_type(2)));
typedef float v8f __attribute__((ext_vector_type(8)));

// -------- workspace layout (float offsets) --------
#define OFF_ALLU 0u                      // 256*17*64   = 278528
#define OFF_ALLN 278528u                 // 278528
#define OFF_UN1  557056u                 // 256*128     = 32768
#define OFF_B96  589824u                 // 96*128      = 12288 (pair-interleaved)
#define OFF_N2   602112u                 // 32
#define OFF_SU   602144u                 // 256*64      = 16384
#define OFF_QU   618528u                 // 256
#define OFF_MU1  618784u                 // 100
#define OFF_RS1  618884u                 // 100
#define OFF_MU2  618984u                 // 100
#define OFF_RS2  619084u                 // 100
#define OFF_H1P  619184u                 // 25600*128   = 3276800
#define OFF_H2P  3895984u                // 3276800
// total ~ 7.17M floats (~28.7 MB)

// ---------------------------------------------------------------------------
// 1) project user / noise fields through lin_w -> all_user / all_noise
// ---------------------------------------------------------------------------
__global__ void k_project(const float* __restrict__ init_user_emb,
                          const float* __restrict__ noise,
                          const float* __restrict__ lin_w,
                          const float* __restrict__ lin_b,
                          float* __restrict__ allu,
                          float* __restrict__ alln) {
    const int f = blockIdx.x;
    const int b = blockIdx.y;
    const int d = threadIdx.x; // 0..63
    const float* src;
    float* dst;
    int sel;
    if (f < NF) {
        sel = (f == 0) ? 0 : 1;
        src = init_user_emb + (size_t)(b * (2 + NUF) + f) * EMB;
        dst = allu + (size_t)(b * NF + f) * DD;
    } else {
        const int f2 = f - NF;
        sel = (f2 == 0) ? 2 : 3;
        src = noise + (size_t)(b * (2 + NUF) + f2) * EMB;
        dst = alln + (size_t)(b * NF + f2) * DD;
    }
    const float* w = lin_w + (size_t)(sel * DD + d) * EMB;
    float acc = lin_b[sel * DD + d];
    #pragma unroll 8
    for (int e = 0; e < EMB; ++e) acc = fmaf(src[e], w[e], acc);
    dst[d] = acc;
}

// ---------------------------------------------------------------------------
// 2) FM user constants: s_user[b][d] = sum_f allu, q_user[b] = sum_f ||e_f||^2
// ---------------------------------------------------------------------------
__global__ void k_fmuser(const float* __restrict__ allu,
                         float* __restrict__ s_user,
                         float* __restrict__ q_user) {
    const int b = blockIdx.x;
    const int d = threadIdx.x; // 0..63
    float s = 0.f, q = 0.f;
    for (int f = 0; f < NF; ++f) {
        float v = allu[(size_t)(b * NF + f) * DD + d];
        s += v;
        q = fmaf(v, v, q);
    }
    s_user[b * DD + d] = s;
    __shared__ float rq[DD];
    rq[d] = q;
    __syncthreads();
    for (int off = DD / 2; off > 0; off >>= 1) {
        if (d < off) rq[d] += rq[d + off];
        __syncthreads();
    }
    if (d == 0) q_user[b] = rq[0];
}

// ---------------------------------------------------------------------------
// 3) per-b H1 contribution of user + noise blocks
// ---------------------------------------------------------------------------
__global__ void k_un1(const float* __restrict__ allu,
                      const float* __restrict__ alln,
                      const float* __restrict__ H1w,
                      const float* __restrict__ H1b,
                      float* __restrict__ UN1) {
    const int b = blockIdx.x;
    const int h = threadIdx.x; // 0..127
    const float* uf = allu + (size_t)b * (NF * DD);
    const float* nf = alln + (size_t)b * (NF * DD);
    const float* w  = H1w + (size_t)h * 4288;
    float acc = H1b[h];
    #pragma unroll 4
    for (int j = 0; j < NF * DD; ++j) {
        acc = fmaf(w[j], uf[j], acc);
        acc = fmaf(w[3200 + j], nf[j], acc);
    }
    UN1[b * HH + h] = acc;
}

// ---------------------------------------------------------------------------
// 4) build B96 in PAIR-INTERLEAVED layout:
//    B96p[((k>>1)*HH + h)*2 + (k&1)] = B[k][h]
//    rows 0..63 = H1w item-emb slice; rows 64..95 = Q[i,h].
//    A WMMA B-fragment (k, k+1 at column h) is then one aligned 8-byte load,
//    contiguous across the 16 lanes of a half-wave.
// ---------------------------------------------------------------------------
__global__ void k_prepB(const float* __restrict__ H1w,
                        const float* __restrict__ W_itemfeat,
                        float* __restrict__ B96p,
                        float* __restrict__ norms2) {
    const int k = blockIdx.x;  // 0..95
    const int h = threadIdx.x; // 0..127
    float v;
    if (k < DD) {
        v = H1w[(size_t)h * 4288 + 1088 + k];
    } else {
        const int i = k - DD;
        const float* wrow = H1w + (size_t)h * 4288 + 1152 + i * DD;
        const float* frow = W_itemfeat + i * DD;
        float acc = 0.f;
        #pragma unroll 8
        for (int d = 0; d < DD; ++d) acc = fmaf(wrow[d], frow[d], acc);
        v = acc;
        if (h == 0) {
            float q = 0.f;
            for (int d = 0; d < DD; ++d) q = fmaf(frow[d], frow[d], q);
            norms2[i] = q;
        }
    }
    B96p[(size_t)((k >> 1) * HH + h) * 2 + (k & 1)] = v;
}

// ---------------------------------------------------------------------------
// 5) H1 GEMM via WMMA: h1_pre[bm,h] = UN1[b,h] + A[bm,0:96] @ B96
//    A row = [W_item[id] (64) | item_feat (32)].  16-row tiles, 4 waves/block,
//    each wave owns 2 column tiles sharing one A fragment per K-step.
// ---------------------------------------------------------------------------
#define AS1 97 // padded LDS stride
__global__ void __launch_bounds__(128) k_h1(const int* __restrict__ item_ids,
                                            const float* __restrict__ W_item,
                                            const float* __restrict__ item_feat,
                                            const float* __restrict__ B96p,
                                            const float* __restrict__ UN1,
                                            float* __restrict__ h1_pre) {
    const int base = blockIdx.x * 16;
    const int t = threadIdx.x;
    __shared__ int   sid[16];
    __shared__ float As[16 * AS1];
    if (t < 16) sid[t] = item_ids[base + t];
    __syncthreads();
    for (int idx = t; idx < 16 * 96; idx += 128) {
        const int r = idx / 96, c = idx % 96;
        float v;
        if (c < DD) v = W_item[(size_t)sid[r] * DD + c];
        else        v = item_feat[(size_t)(base + r) * NIF + (c - DD)];
        As[r * AS1 + c] = v;
    }
    __syncthreads();

    const int wave  = t >> 5;
    const int lane  = t & 31;
    const int lrow  = lane & 15;
    const int khalf = lane >> 4;
    const int hb0   = wave * 32;       // tiles hb0 and hb0+16

    v8f acc0 = {0.f, 0.f, 0.f, 0.f, 0.f, 0.f, 0.f, 0.f};
    v8f acc1 = acc0;
    #pragma unroll
    for (int k0 = 0; k0 < 96; k0 += 4) {
        const int ka = k0 + 2 * khalf;       // even
        v2f a;
        a.x = As[lrow * AS1 + ka];
        a.y = As[lrow * AS1 + ka + 1];
        const float* bp = B96p + (size_t)((ka >> 1) * HH) * 2;
        v2f b0 = *(const v2f*)(bp + (size_t)(hb0 + lrow) * 2);
        v2f b1 = *(const v2f*)(bp + (size_t)(hb0 + 16 + lrow) * 2);
        acc0 = __builtin_amdgcn_wmma_f32_16x16x4_f32(
            false, a, false, b0, (short)0, acc0, false, false);
        acc1 = __builtin_amdgcn_wmma_f32_16x16x4_f32(
            false, a, false, b1, (short)0, acc1, false, false);
    }
    #pragma unroll
    for (int v = 0; v < 8; ++v) {
        const int row = v + 8 * khalf;
        const int bm  = base + row;
        const int bidx = bm / MM;
        const int col0 = hb0 + lrow;
        h1_pre[(size_t)bm * HH + col0]      = acc0[v] + UN1[bidx * HH + col0];
        h1_pre[(size_t)bm * HH + col0 + 16] = acc1[v] + UN1[bidx * HH + col0 + 16];
    }
}

// ---------------------------------------------------------------------------
// 6) BatchNorm stats over axes (b,h) per channel m (biased var, eps 1e-5)
// ---------------------------------------------------------------------------
__global__ void k_stats(const float* __restrict__ x,
                        float* __restrict__ mu,
                        float* __restrict__ rstd) {
    const int m = blockIdx.x;
    const int t = threadIdx.x; // 256 threads
    float s = 0.f, q = 0.f;
    for (int idx = t; idx < BZ * HH; idx += 256) {
        const int b = idx >> 7;
        const int h = idx & 127;
        float v = x[(size_t)(b * MM + m) * HH + h];
        s += v;
        q = fmaf(v, v, q);
    }
    __shared__ float rs[256], rq[256];
    rs[t] = s; rq[t] = q;
    __syncthreads();
    for (int off = 128; off > 0; off >>= 1) {
        if (t < off) { rs[t] += rs[t + off]; rq[t] += rq[t + off]; }
        __syncthreads();
    }
    if (t == 0) {
        const float inv = 1.f / (float)(BZ * HH);
        float mean = rs[0] * inv;
        float var  = rq[0] * inv - mean * mean;
        mu[m]   = mean;
        rstd[m] = rsqrtf(var + 1e-5f);
    }
}

// ---------------------------------------------------------------------------
// 7) H2 GEMM via WMMA with fused BN1+ReLU on the A side:
//    h2_pre[bm,g] = relu(bn1(h1_pre[bm,:])) @ H2w[g,:] + H2b[g]
//    B fragment (H2w[g][ka], [ka+1]) is contiguous -> single b64 load.
// ---------------------------------------------------------------------------
#define AS2 129
__global__ void __launch_bounds__(128) k_h2(const float* __restrict__ h1_pre,
                                            const float* __restrict__ mu1,
                                            const float* __restrict__ rs1,
                                            const float* __restrict__ bn1g,
                                            const float* __restrict__ bn1b,
                                            const float* __restrict__ H2w,
                                            const float* __restrict__ H2b,
                                            float* __restrict__ h2_pre) {
    const int base = blockIdx.x * 16;
    const int t = threadIdx.x;
    __shared__ float As[16 * AS2];
    for (int idx = t; idx < 16 * HH; idx += 128) {
        const int r = idx >> 7, h = idx & 127;
        const int bm = base + r;
        const int m  = bm % MM;
        float x  = h1_pre[(size_t)bm * HH + h];
        float xn = fmaf(bn1g[m] * rs1[m], x - mu1[m], bn1b[m]);
        As[r * AS2 + h] = fmaxf(xn, 0.f);
    }
    __syncthreads();

    const int wave  = t >> 5;
    const int lane  = t & 31;
    const int lrow  = lane & 15;
    const int khalf = lane >> 4;
    const int gb0   = wave * 32;       // tiles gb0 and gb0+16

    v8f acc0 = {0.f, 0.f, 0.f, 0.f, 0.f, 0.f, 0.f, 0.f};
    v8f acc1 = acc0;
    #pragma unroll
    for (int k0 = 0; k0 < HH; k0 += 4) {
        const int ka = k0 + 2 * khalf;       // even
        v2f a;
        a.x = As[lrow * AS2 + ka];
        a.y = As[lrow * AS2 + ka + 1];
        // B[k][g] = H2w[g][k]  (einsum 'bmh,gh->bmg'); pair contiguous in k
        v2f b0 = *(const v2f*)(H2w + (size_t)(gb0 + lrow) * HH + ka);
        v2f b1 = *(const v2f*)(H2w + (size_t)(gb0 + 16 + lrow) * HH + ka);
        acc0 = __builtin_amdgcn_wmma_f32_16x16x4_f32(
            false, a, false, b0, (short)0, acc0, false, false);
        acc1 = __builtin_amdgcn_wmma_f32_16x16x4_f32(
            false, a, false, b1, (short)0, acc1, false, false);
    }
    #pragma unroll
    for (int v = 0; v < 8; ++v) {
        const int row = v + 8 * khalf;
        const int bm  = base + row;
        const int col0 = gb0 + lrow;
        h2_pre[(size_t)bm * HH + col0]      = acc0[v] + H2b[col0];
        h2_pre[(size_t)bm * HH + col0 + 16] = acc1[v] + H2b[col0 + 16];
    }
}

// ---------------------------------------------------------------------------
// 8) final fuse: BN2+ReLU -> H3 dot; FM; ratings; bias.  1 block per (b,m).
// ---------------------------------------------------------------------------
__global__ void __launch_bounds__(128) k_final(
    const float* __restrict__ h2_pre,
    const float* __restrict__ mu2, const float* __restrict__ rs2,
    const float* __restrict__ bn2g, const float* __restrict__ bn2b,
    const float* __restrict__ H3w, const float* __restrict__ H3b,
    const int* __restrict__ item_ids,
    const float* __restrict__ W_item,
    const float* __restrict__ item_feat,
    const float* __restrict__ W_itemfeat,
    const float* __restrict__ s_user,
    const float* __restrict__ q_user,
    const float* __restrict__ norms2,
    const float* __restrict__ item_bias,
    const float* __restrict__ init_user_emb,
    const float* __restrict__ ratings,
    const float* __restrict__ p1, const float* __restrict__ p2,
    const float* __restrict__ p3, const float* __restrict__ p4,
    float* __restrict__ out) {
    const int bm = blockIdx.x;
    const int b  = bm / MM;
    const int m  = bm % MM;
    const int t  = threadIdx.x; // 128 threads
    const int id = item_ids[bm];

    __shared__ float sfeat[NIF];
    __shared__ float redA[128];
    __shared__ float redB[128];
    if (t < NIF) sfeat[t] = item_feat[(size_t)bm * NIF + t];
    __syncthreads();

    // linear head partial (all 128 threads)
    float x  = h2_pre[(size_t)bm * HH + t];
    float xn = fmaf(bn2g[m] * rs2[m], x - mu2[m], bn2b[m]);
    float lin_p = fmaxf(xn, 0.f) * H3w[t];

    // FM partials
    float fm_p = 0.f;
    if (t < DD) {
        float w = W_item[(size_t)id * DD + t];
        float s = s_user[b * DD + t] + w;
        #pragma unroll 8
        for (int i = 0; i < NIF; ++i) s = fmaf(sfeat[i], W_itemfeat[i * DD + t], s);
        fm_p = fmaf(s, s, -w * w);
    }
    if (t < NIF) {
        float f = sfeat[t];
        fm_p -= f * f * norms2[t];
    }

    redA[t] = lin_p;
    redB[t] = fm_p;
    __syncthreads();
    for (int off = 64; off > 0; off >>= 1) {
        if (t < off) { redA[t] += redA[t + off]; redB[t] += redB[t + off]; }
        __syncthreads();
    }
    if (t == 0) {
        const float lin = redA[0] + H3b[0];
        const float fm  = 0.5f * (redB[0] - q_user[b]);
        const float ub  = init_user_emb[(size_t)(b * (2 + NUF) + (1 + NUF)) * EMB];
        out[bm] = p1[0] * fm + p2[0] * lin + p3[0] * ratings[bm] +
                  p4[0] * (item_bias[id] + ub);
    }
}

// ---------------------------------------------------------------------------
extern "C" void kernel_launch(void* const* d_in, const int* in_sizes, int n_in,
                              void* d_out, int out_size, void* d_ws, size_t ws_size,
                              hipStream_t stream) {
    const float* ratings       = (const float*)d_in[0];
    const int*   item_ids      = (const int*)  d_in[1];
    const float* noise         = (const float*)d_in[2];
    const float* init_user_emb = (const float*)d_in[3];
    const float* item_feat     = (const float*)d_in[4];
    const float* W_item        = (const float*)d_in[5];
    const float* item_bias     = (const float*)d_in[6];
    const float* W_itemfeat    = (const float*)d_in[7];
    const float* lin_w         = (const float*)d_in[8];
    const float* lin_b         = (const float*)d_in[9];
    const float* H1w           = (const float*)d_in[10];
    const float* H1b           = (const float*)d_in[11];
    const float* bn1g          = (const float*)d_in[12];
    const float* bn1b          = (const float*)d_in[13];
    const float* H2w           = (const float*)d_in[14];
    const float* H2b           = (const float*)d_in[15];
    const float* bn2g          = (const float*)d_in[16];
    const float* bn2b          = (const float*)d_in[17];
    const float* H3w           = (const float*)d_in[18];
    const float* H3b           = (const float*)d_in[19];
    const float* p1            = (const float*)d_in[20];
    const float* p2            = (const float*)d_in[21];
    const float* p3            = (const float*)d_in[22];
    const float* p4            = (const float*)d_in[23];
    float* out = (float*)d_out;
    float* ws  = (float*)d_ws;

    float* allu   = ws + OFF_ALLU;
    float* alln   = ws + OFF_ALLN;
    float* UN1    = ws + OFF_UN1;
    float* B96p   = ws + OFF_B96;
    float* norms2 = ws + OFF_N2;
    float* s_user = ws + OFF_SU;
    float* q_user = ws + OFF_QU;
    float* mu1    = ws + OFF_MU1;
    float* rs1    = ws + OFF_RS1;
    float* mu2    = ws + OFF_MU2;
    float* rs2    = ws + OFF_RS2;
    float* h1p    = ws + OFF_H1P;
    float* h2p    = ws + OFF_H2P;

    k_project<<<dim3(2 * NF, BZ), DD, 0, stream>>>(init_user_emb, noise, lin_w,
                                                   lin_b, allu, alln);
    k_fmuser<<<BZ, DD, 0, stream>>>(allu, s_user, q_user);
    k_un1<<<BZ, HH, 0, stream>>>(allu, alln, H1w, H1b, UN1);
    k_prepB<<<96, HH, 0, stream>>>(H1w, W_itemfeat, B96p, norms2);
    k_h1<<<ROWS / 16, 128, 0, stream>>>(item_ids, W_item, item_feat, B96p, UN1, h1p);
    k_stats<<<MM, 256, 0, stream>>>(h1p, mu1, rs1);
    k_h2<<<ROWS / 16, 128, 0, stream>>>(h1p, mu1, rs1, bn1g, bn1b, H2w, H2b, h2p);
    k_stats<<<MM, 256, 0, stream>>>(h2p, mu2, rs2);
    k_final<<<ROWS, 128, 0, stream>>>(h2p, mu2, rs2, bn2g, bn2b, H3w, H3b,
                                      item_ids, W_item, item_feat, W_itemfeat,
                                      s_user, q_user, norms2, item_bias,
                                      init_user_emb, ratings, p1, p2, p3, p4, out);
}